// SumReadout_63986422776374
// MI455X (gfx1250) — compile-verified
//
#include <hip/hip_runtime.h>
#include <hip/hip_bf16.h>

typedef __attribute__((ext_vector_type(2))) float v2f;
typedef __attribute__((ext_vector_type(8))) float v8f;
typedef __attribute__((ext_vector_type(4))) int   v4i;

#define D_FEAT    128
#define ROWS      64
#define THREADS   256
#define ROW_PITCH 132   // padded LDS row pitch (floats): avoids 2-way bank conflicts

#if defined(__has_builtin)
# if __has_builtin(__builtin_amdgcn_global_load_async_to_lds_b128)
#  define HAVE_ASYNC_LDS 1
# endif
# if __has_builtin(__builtin_amdgcn_s_wait_asynccnt)
#  define HAVE_WAIT_ASYNC_BUILTIN 1
# endif
#endif

typedef __attribute__((address_space(3))) v4i* lds_v4i_p;

static __device__ __forceinline__ void wait_asynccnt0() {
#ifdef HAVE_WAIT_ASYNC_BUILTIN
    __builtin_amdgcn_s_wait_asynccnt(0);
#else
    asm volatile("s_wait_asynccnt 0x0" ::: "memory");
#endif
}

// Zero the output accumulator (harness poisons d_out with 0xAA).
__global__ void SumReadout_zero_kernel(float* __restrict__ out, int n) {
    int i = blockIdx.x * blockDim.x + threadIdx.x;
    if (i < n) out[i] = 0.0f;
}

// Segment-sum over sorted node2graph using one-hot WMMA reduction.
// Block = 64 contiguous rows; wave w owns features [16w,16w+16).
// Tile staged global->LDS with async b128 copies (ASYNCcnt), overlapped with
// the segment-id reduction; then D(16x16) = A(one-hot) x B(feature tile) + C
// on the matrix pipe, flushed with global atomics at segment granularity.
__global__ __launch_bounds__(THREADS)
void SumReadout_63986422776374_kernel(const float* __restrict__ x,
                                      const int* __restrict__ seg,
                                      float* __restrict__ out,
                                      int n_nodes) {
    __shared__ float s_x[ROWS * ROW_PITCH];   // 33 KB padded tile
    __shared__ int   s_seg[ROWS];
    __shared__ int   s_max_local;

    const int  tid  = threadIdx.x;
    const int  r0   = blockIdx.x * ROWS;
    const bool full = (r0 + ROWS) <= n_nodes;   // block-uniform

    // ---- 1) Kick off the async tile copy first (latency hiding) ----
    if (full) {
        const float* gsrc = x + (size_t)r0 * D_FEAT;
#ifdef HAVE_ASYNC_LDS
        #pragma unroll
        for (int it = 0; it < (ROWS * D_FEAT / 4) / THREADS; ++it) {
            const int j   = it * THREADS + tid;   // 16-byte transfer index
            const int row = j >> 5;               // 32 transfers per 128-f row
            const int c4  = (j & 31) << 2;
            __builtin_amdgcn_global_load_async_to_lds_b128(
                (v4i*)(gsrc + row * D_FEAT + c4),
                (lds_v4i_p)(&s_x[row * ROW_PITCH + c4]),
                0, 0);
        }
#else
        #pragma unroll
        for (int it = 0; it < (ROWS * D_FEAT / 4) / THREADS; ++it) {
            const int j   = it * THREADS + tid;
            const int row = j >> 5;
            const int c4  = (j & 31) << 2;
            const float4 v = *(const float4*)(gsrc + row * D_FEAT + c4);
            *(float4*)(&s_x[row * ROW_PITCH + c4]) = v;
        }
#endif
    }

    // ---- 2) Stage segment ids + local-span reduction while copy is in flight ----
    if (tid == 0) s_max_local = 0;
    if (tid < ROWS) {
        const int r = r0 + tid;
        s_seg[tid] = (r < n_nodes) ? seg[r] : 0;
    }
    __syncthreads();

    const int seg_first = s_seg[0];
    if (tid < ROWS && (r0 + tid) < n_nodes)
        atomicMax(&s_max_local, s_seg[tid] - seg_first);

#ifdef HAVE_ASYNC_LDS
    if (full) wait_asynccnt0();   // each wave drains its own async copies
#endif
    __syncthreads();              // tile + s_max_local visible block-wide

    const int max_local = s_max_local;   // block-uniform

    if (full && max_local < 16) {
        // ---- 3) WMMA path (EXEC all-ones: branch is block-uniform) ----
        const int lane  = tid & 31;
        const int wave  = tid >> 5;
        const int mn    = lane & 15;            // A row (m) == B col (n)
        const int khalf = (lane < 16) ? 0 : 2;  // K pair of this half-wave
        const int fbase = wave * 16;            // feature group of this wave

        v8f acc = {};
        #pragma unroll 4
        for (int k4 = 0; k4 < ROWS / 4; ++k4) {
            const int rk = k4 * 4 + khalf;
            v2f a, b;
            // A: one-hot local-segment selector (VGPR0=K, VGPR1=K+1)
            a.x = (s_seg[rk]     - seg_first == mn) ? 1.0f : 0.0f;
            a.y = (s_seg[rk + 1] - seg_first == mn) ? 1.0f : 0.0f;
            // B: feature tile rows from LDS (padded, conflict-free pitch)
            b.x = s_x[(rk)     * ROW_PITCH + fbase + mn];
            b.y = s_x[(rk + 1) * ROW_PITCH + fbase + mn];
            // (neg_a, A, neg_b, B, c_mod, C, reuse_a, reuse_b)
            acc = __builtin_amdgcn_wmma_f32_16x16x4_f32(
                false, a, false, b, (short)0, acc, false, false);
        }

        // Flush: acc VGPR r holds D[m=r] (lanes 0-15) / D[m=8+r] (lanes 16-31)
        const int mbase = (lane < 16) ? 0 : 8;
        #pragma unroll
        for (int r = 0; r < 8; ++r) {
            const int mm = mbase + r;
            if (mm <= max_local) {
                atomicAdd(&out[(size_t)(seg_first + mm) * D_FEAT + fbase + mn],
                          acc[r]);
            }
        }
    } else {
        // ---- Fallback: >16 segments in chunk, or grid tail ----
        for (int idx = tid; idx < ROWS * D_FEAT; idx += THREADS) {
            const int rr = idx >> 7;
            const int r  = r0 + rr;
            if (r < n_nodes) {
                const int f = idx & (D_FEAT - 1);
                const float v = full ? s_x[rr * ROW_PITCH + f]
                                     : x[(size_t)r * D_FEAT + f];
                atomicAdd(&out[(size_t)s_seg[rr] * D_FEAT + f], v);
            }
        }
    }
}

extern "C" void kernel_launch(void* const* d_in, const int* in_sizes, int n_in,
                              void* d_out, int out_size, void* d_ws, size_t ws_size,
                              hipStream_t stream) {
    const float* x   = (const float*)d_in[0];   // [N_NODES, D_FEAT] f32
    const int*   seg = (const int*)d_in[1];     // [N_NODES] sorted i32
    float*       out = (float*)d_out;           // [BATCH, D_FEAT] f32

    const int n_nodes = in_sizes[0] / D_FEAT;

    // 1) Zero the accumulator output.
    SumReadout_zero_kernel<<<(out_size + THREADS - 1) / THREADS, THREADS, 0, stream>>>(
        out, out_size);

    // 2) Chunked one-hot WMMA segment reduction with async LDS staging.
    const int nblocks = (n_nodes + ROWS - 1) / ROWS;
    SumReadout_63986422776374_kernel<<<nblocks, THREADS, 0, stream>>>(
        x, seg, out, n_nodes);
}